// SPNNeuronShallow_8478265442320
// MI455X (gfx1250) — compile-verified
//
#include <hip/hip_runtime.h>
#include <cstdint>

// SPN shallow Gaussian mixture layer, MI455X (gfx1250, wave32).
// Memory floor: 64 MB x-traffic @ 23.3 TB/s => ~2.8us. Previous rev was
// VALU-bound (~84 slots/row/wave); this rev halves math slots via CDNA5
// packed-f32 VOP3P (v_pk_fma_f32/v_pk_add_f32) by pairing each lane's two
// groups into <2 x float>, and removes ln<->log2 conversion multiplies by
// pre-scaling folded params into the native v_exp_f32/v_log_f32 base-2 domain.
// Data path: per-wave async global->LDS (ASYNCcnt) software pipeline with
// in-order completion waits; cross-lane reduce on DPP8 + permlane16/x16.
// WMMA intentionally unused: weights are block-diagonal (4 nonzeros per
// 256-long column); dense f32 16x16x4 WMMA inflates FLOPs ~64x past HBM floor.

#define NFEAT   128
#define NMULT   3
#define THREADS 256
#define ROWS_PER_BLOCK 256
#define HALF_LOG_2PI 0.9189385332046727f
#define LOG2E_F 1.4426950408889634f
#define LN2_F   0.6931471805599453f

typedef float v2f __attribute__((ext_vector_type(2)));

__device__ __forceinline__ float exp2_fast(float x) {
#if __has_builtin(__builtin_amdgcn_exp2f)
    return __builtin_amdgcn_exp2f(x);       // v_exp_f32 (native base-2)
#else
    return exp2f(x);
#endif
}
__device__ __forceinline__ float log2_fast(float x) {
#if __has_builtin(__builtin_amdgcn_logf)
    return __builtin_amdgcn_logf(x);        // v_log_f32 (native base-2)
#else
    return log2f(x);
#endif
}

__device__ __forceinline__ constexpr int dpp8sel(int a, int b, int c, int d,
                                                 int e, int f, int g, int h) {
    return a | (b << 3) | (c << 6) | (d << 9) | (e << 12) | (f << 15) |
           (g << 18) | (h << 21);
}

__device__ __forceinline__ float wave32_allreduce_add(float p) {
#if __has_builtin(__builtin_amdgcn_mov_dpp8) && \
    __has_builtin(__builtin_amdgcn_permlane16) && \
    __has_builtin(__builtin_amdgcn_permlanex16)
    p += __int_as_float(__builtin_amdgcn_mov_dpp8(
             __float_as_int(p), dpp8sel(1, 0, 3, 2, 5, 4, 7, 6)));   // xor1
    p += __int_as_float(__builtin_amdgcn_mov_dpp8(
             __float_as_int(p), dpp8sel(2, 3, 0, 1, 6, 7, 4, 5)));   // xor2
    p += __int_as_float(__builtin_amdgcn_mov_dpp8(
             __float_as_int(p), dpp8sel(4, 5, 6, 7, 0, 1, 2, 3)));   // xor4
    p += __int_as_float(__builtin_amdgcn_permlane16(                 // xor8
             __float_as_int(p), __float_as_int(p),
             0xFEDCBA98u, 0x76543210u, false, false));
    p += __int_as_float(__builtin_amdgcn_permlanex16(                // xor16
             __float_as_int(p), __float_as_int(p),
             0x76543210u, 0xFEDCBA98u, false, false));
#else
    #pragma unroll
    for (int off = 16; off > 0; off >>= 1) p += __shfl_xor(p, off, 32);
#endif
    return p;
}

__global__ __launch_bounds__(THREADS, 2)
void spn_shallow_kernel(const float* __restrict__ x,
                        const int*   __restrict__ perm,
                        const float* __restrict__ means,
                        const float* __restrict__ stds,
                        const float* __restrict__ sumw,
                        float* __restrict__ out,
                        int rows)
{
    __shared__ float tile[ROWS_PER_BLOCK * NFEAT];   // 128 KB, row-major

    const int tid  = threadIdx.x;
    const int lane = tid & 31;
    const int wave = tid >> 5;
    const int wrow0 = blockIdx.x * ROWS_PER_BLOCK + wave * 32; // wave's 32 rows

    // -------- per-wave async staging: 1 b128 chunk per lane per row ---------
    // Single shared base; the 24-bit instruction offset applies to BOTH the
    // LDS destination and the global source -> zero per-issue VALU.
    float* wtile = &tile[(wave * 32) * NFEAT];
    {
        const unsigned ldsa = (unsigned)(uintptr_t)wtile + (unsigned)lane * 16u;
        const uint64_t ga   = (uint64_t)(uintptr_t)(x + (size_t)wrow0 * NFEAT) +
                              (uint64_t)lane * 16u;
        #pragma unroll
        for (int j = 0; j < 32; ++j) {
            asm volatile("global_load_async_to_lds_b128 %0, %1, off offset:%2"
                         :: "v"(ldsa), "v"(ga), "i"(j * (NFEAT * 4)) : "memory");
        }
    }

    // ---- per-lane folded params, packed per feature-slot across 2 groups ---
    // ll(x) = a*x^2 + b*x + c ; a=-1/(2s^2), b=mu/s^2,
    // c=-mu^2/(2s^2)-log(s)-HALF_LOG_2PI ; D[m]=c_pair[m]+logsoftmax_m(w)
    // All of A,B,D pre-scaled by log2(e): whole hot loop runs in base-2 so
    // v_exp_f32/v_log_f32 need no conversion multiplies; one *ln2 at store.
    v2f A2[2][NMULT], B2[2][NMULT], D2[NMULT];   // component gg = group pair
    int col[2][2];
    #pragma unroll
    for (int gg = 0; gg < 2; ++gg) {
        const int g = lane + gg * 32;
        float csum[NMULT];
        #pragma unroll
        for (int m = 0; m < NMULT; ++m) csum[m] = -2.0f * HALF_LOG_2PI;
        #pragma unroll
        for (int s = 0; s < 2; ++s) {
            const int f = 2 * g + s;
            col[gg][s] = perm[f];
            #pragma unroll
            for (int m = 0; m < NMULT; ++m) {
                const float mu   = means[f * NMULT + m];
                const float sd   = stds [f * NMULT + m];
                const float inv  = 1.0f / sd;
                const float inv2 = inv * inv;
                A2[s][m][gg] = -0.5f * inv2 * LOG2E_F;
                B2[s][m][gg] = mu * inv2 * LOG2E_F;
                csum[m] += -0.5f * mu * mu * inv2 - __logf(sd);
            }
        }
        const float w0 = sumw[g * NMULT + 0];
        const float w1 = sumw[g * NMULT + 1];
        const float w2 = sumw[g * NMULT + 2];
        const float wm = fmaxf(w0, fmaxf(w1, w2));
        const float lw = wm + __logf(__expf(w0 - wm) + __expf(w1 - wm) +
                                     __expf(w2 - wm));
        D2[0][gg] = (csum[0] + (w0 - lw)) * LOG2E_F;
        D2[1][gg] = (csum[1] + (w1 - lw)) * LOG2E_F;
        D2[2][gg] = (csum[2] + (w2 - lw)) * LOG2E_F;
    }

    // -------- pipelined compute: wait on in-order completions batch-wise ----
    float myout = 0.0f;
    #pragma unroll
    for (int bat = 0; bat < 4; ++bat) {
        // Async loads complete in issue order -> cnt<=24-8*bat guarantees
        // rows [0, 8*(bat+1)) of this wave are resident in LDS.
        asm volatile("s_wait_asynccnt %0" :: "i"(24 - bat * 8) : "memory");
        for (int jj = 0; jj < 8; ++jj) {
            const int j = bat * 8 + jj;
            const float* row = &wtile[j * NFEAT];
            // Pack the two groups' inputs: X0/X1 = {slot0,slot1} per group pair
            v2f X0, X1;
            X0.x = row[col[0][0]];  X0.y = row[col[1][0]];
            X1.x = row[col[0][1]];  X1.y = row[col[1][1]];
            v2f T[NMULT];
            #pragma unroll
            for (int m = 0; m < NMULT; ++m) {            // v_pk_fma_f32 x4
                const v2f U0 = __builtin_elementwise_fma(A2[0][m], X0, B2[0][m]);
                const v2f U1 = __builtin_elementwise_fma(A2[1][m], X1, B2[1][m]);
                T[m] = __builtin_elementwise_fma(
                           U0, X0, __builtin_elementwise_fma(U1, X1, D2[m]));
            }
            v2f MX;                                      // v_max3_num_f32 x2
            MX.x = fmaxf(fmaxf(T[0].x, T[1].x), T[2].x);
            MX.y = fmaxf(fmaxf(T[0].y, T[1].y), T[2].y);
            const v2f d0 = T[0] - MX, d1 = T[1] - MX, d2 = T[2] - MX; // pk_add
            const float esx = exp2_fast(d0.x) + exp2_fast(d1.x) + exp2_fast(d2.x);
            const float esy = exp2_fast(d0.y) + exp2_fast(d1.y) + exp2_fast(d2.y);
            float p = (MX.x + log2_fast(esx)) + (MX.y + log2_fast(esy));
            p = wave32_allreduce_add(p);                 // VALU-pipe butterfly
            if (lane == j) myout = p;                    // compact row j
        }
    }

    const int orow = wrow0 + lane;
    if (orow < rows) out[orow] = myout * LN2_F;  // back to natural log; coalesced
}

extern "C" void kernel_launch(void* const* d_in, const int* in_sizes, int n_in,
                              void* d_out, int out_size, void* d_ws, size_t ws_size,
                              hipStream_t stream) {
    const float* x     = (const float*)d_in[0];
    const int*   perm  = (const int*)  d_in[1];
    const float* means = (const float*)d_in[2];
    const float* stds  = (const float*)d_in[3];
    const float* sumw  = (const float*)d_in[4];
    float*       out   = (float*)d_out;

    const int rows = in_sizes[0] / NFEAT;                           // 131072
    const int grid = (rows + ROWS_PER_BLOCK - 1) / ROWS_PER_BLOCK;  // 512
    spn_shallow_kernel<<<grid, THREADS, 0, stream>>>(x, perm, means, stds, sumw,
                                                     out, rows);
}